// LiftedStructureLoss_80693845557677
// MI455X (gfx1250) — compile-verified
//
#include <hip/hip_runtime.h>
#include <hip/hip_bf16.h>

typedef __attribute__((ext_vector_type(2))) float v2f;
typedef __attribute__((ext_vector_type(8))) float v8f;

#define MARGIN 1.0f

// ---------------------------------------------------------------------------
// V_WMMA_F32_16X16X4_F32 fragment mapping (per ISA layout tables):
// A (16x4 MxK):  lane -> row = rowbase + (lane&15), k = kb + 2*(lane>=16) .. +1
// B (4x16 KxN):  lane -> col = colbase + (lane&15), same k formula
// C/D:           c[r] = G[rowbase + r + 8*(lane>=16)][colbase + (lane&15)]
// ---------------------------------------------------------------------------

__device__ __forceinline__ v8f wmma_f32(v2f a, v2f b, v8f c) {
    return __builtin_amdgcn_wmma_f32_16x16x4_f32(false, a, false, b,
                                                 (short)0, c, false, false);
}

template <int KS>
__device__ __forceinline__ void load_frag(v2f (&b)[KS], const float* __restrict__ p) {
    #pragma unroll
    for (int s = 0; s < KS; ++s) b[s] = *(const v2f*)(p + 4 * s);
}

template <int KS>
__device__ __forceinline__ v8f mma_chain(const v2f (&a)[KS], const v2f (&b)[KS]) {
    v8f c = {};
    #pragma unroll
    for (int s = 0; s < KS; ++s) c = wmma_f32(a[s], b[s], c);
    return c;
}

// Generic (runtime-D) Gram tile: load+MMA interleaved (fallback path).
__device__ inline v8f gram_tile_gen(const float* __restrict__ X, int d,
                                    int rowbase, int colbase) {
    const int lane = threadIdx.x & 31;
    const int l15  = lane & 15;
    const int k0   = (lane >> 4) << 1;
    const float* ap = X + (size_t)(rowbase + l15) * d + k0;
    const float* bp = X + (size_t)(colbase + l15) * d + k0;
    v8f c = {};
    #pragma unroll 8
    for (int kb = 0; kb < d; kb += 4) {
        v2f a = *(const v2f*)(ap + kb);
        v2f b = *(const v2f*)(bp + kb);
        c = wmma_f32(a, b, c);
    }
    return c;
}

// ---------------------------------------------------------------------------
// Workspace init: zero row_neg, scalar sum, pair count.
// ---------------------------------------------------------------------------
__global__ void init_ws_kernel(float* __restrict__ row_neg,
                               float* __restrict__ sum,
                               unsigned* __restrict__ cnt, int n) {
    int i = blockIdx.x * blockDim.x + threadIdx.x;
    if (i < n) row_neg[i] = 0.0f;
    if (i == 0) { sum[0] = 0.0f; cnt[0] = 0u; }
}

// ---------------------------------------------------------------------------
// sq[i] = ||x_i||^2, one wave32 per row.
// ---------------------------------------------------------------------------
__global__ __launch_bounds__(32) void sqnorm_kernel(const float* __restrict__ X,
                                                    float* __restrict__ sq, int d) {
    const int row  = blockIdx.x;
    const int lane = threadIdx.x;
    const float* p = X + (size_t)row * d;
    float s = 0.0f;
    for (int k = lane; k < d; k += 32) { float v = p[k]; s += v * v; }
    #pragma unroll
    for (int m = 1; m < 32; m <<= 1) s += __shfl_xor(s, m, 32);
    if (lane == 0) sq[row] = s;
}

// ---------------------------------------------------------------------------
// len_p = #(i,j): i!=j, t_i == t_j   (full pos_mask count)
// ---------------------------------------------------------------------------
__global__ void count_pos_kernel(const int* __restrict__ tgt, int n,
                                 unsigned* __restrict__ cnt) {
    int i = blockIdx.x * blockDim.x + threadIdx.x;
    if (i >= n) return;
    const int t = tgt[i];
    unsigned c = 0;
    for (int j = 0; j < n; ++j) c += (tgt[j] == t) ? 1u : 0u;
    atomicAdd(cnt, c - 1u);  // remove diagonal
}

// ===========================================================================
// Pass 1 fast path (compile-time D): A-strip resident in registers for the
// wave's lifetime; B fragments double-buffered so the loads for tile ct+1
// overlap the 32-deep WMMA chain for tile ct (explicit 2x unroll keeps all
// fragment indices static -> pure register traffic, no scratch).
// ===========================================================================
template <int D>
__global__ __launch_bounds__(32) void pass1_rowneg_fast(
        const float* __restrict__ X, const float* __restrict__ sq,
        const int* __restrict__ tgt, float* __restrict__ row_neg,
        int ntiles, int tilesPerChunk) {
    constexpr int KS = D / 4;
    const int rowbase = blockIdx.x * 16;
    const int lane    = threadIdx.x;
    const int l15     = lane & 15;
    const int halfsel = lane >> 4;
    const int k0      = halfsel << 1;

    const int ct0 = blockIdx.y * tilesPerChunk;
    int ct1 = ct0 + tilesPerChunk; if (ct1 > ntiles) ct1 = ntiles;

    // Preload entire A strip (16 rows x D) into registers: KS v2f per lane.
    v2f afrag[KS];
    load_frag<KS>(afrag, X + (size_t)(rowbase + l15) * D + k0);

    float sqr[8]; int tr[8];
    #pragma unroll
    for (int r = 0; r < 8; ++r) {
        const int row = rowbase + r + 8 * halfsel;
        sqr[r] = sq[row];
        tr[r]  = tgt[row];
    }

    float acc[8] = {0.f, 0.f, 0.f, 0.f, 0.f, 0.f, 0.f, 0.f};

    const float* colX = X + (size_t)l15 * D + k0;  // + colbase*D per tile

    auto epilogue = [&](const v8f& c, int ct) {
        const int colbase = ct * 16;
        const float sqc = sq[colbase + l15];
        const int   tc  = tgt[colbase + l15];
        #pragma unroll
        for (int r = 0; r < 8; ++r) {
            float d2   = fmaxf(sqr[r] + sqc - 2.0f * c[r], 0.0f);
            float dist = sqrtf(d2);
            if (tr[r] != tc && dist != 0.0f)
                acc[r] += __expf(MARGIN - dist);
        }
    };

    if (ct0 < ct1) {
        v2f b0[KS], b1[KS];
        int ct = ct0;
        load_frag<KS>(b0, colX + (size_t)ct * 16 * D);
        while (ct + 2 <= ct1) {
            load_frag<KS>(b1, colX + (size_t)(ct + 1) * 16 * D);   // prefetch ct+1
            epilogue(mma_chain<KS>(afrag, b0), ct);                // compute ct
            if (ct + 2 < ct1)
                load_frag<KS>(b0, colX + (size_t)(ct + 2) * 16 * D); // prefetch ct+2
            epilogue(mma_chain<KS>(afrag, b1), ct + 1);            // compute ct+1
            ct += 2;
        }
        if (ct < ct1)                                              // odd tail in b0
            epilogue(mma_chain<KS>(afrag, b0), ct);
    }

    // reduce across the 16 lanes of each half (halves hold disjoint rows)
    #pragma unroll
    for (int r = 0; r < 8; ++r) {
        float v = acc[r];
        v += __shfl_xor(v, 1, 32);
        v += __shfl_xor(v, 2, 32);
        v += __shfl_xor(v, 4, 32);
        v += __shfl_xor(v, 8, 32);
        if (l15 == 0) atomicAdd(&row_neg[rowbase + r + 8 * halfsel], v);
    }
}

// ===========================================================================
// Pass 2 fast path: one upper-triangular tile per wave.
// ===========================================================================
template <int D>
__global__ __launch_bounds__(32) void pass2_loss_fast(
        const float* __restrict__ X, const float* __restrict__ sq,
        const int* __restrict__ tgt, const float* __restrict__ row_neg,
        float* __restrict__ out_sum) {
    constexpr int KS = D / 4;
    const int rowTile = blockIdx.x;
    const int colTile = blockIdx.y;
    if (colTile < rowTile) return;   // wave-uniform skip of strictly-lower tiles
    const int rowbase = rowTile * 16;
    const int colbase = colTile * 16;
    const int lane    = threadIdx.x;
    const int l15     = lane & 15;
    const int halfsel = lane >> 4;
    const int k0      = halfsel << 1;

    v2f afrag[KS], bfrag[KS];
    load_frag<KS>(afrag, X + (size_t)(rowbase + l15) * D + k0);
    load_frag<KS>(bfrag, X + (size_t)(colbase + l15) * D + k0);
    v8f c = mma_chain<KS>(afrag, bfrag);

    const int   col = colbase + l15;
    const float sqc = sq[col];
    const int   tc  = tgt[col];
    const float rnc = row_neg[col];

    float s = 0.0f;
    #pragma unroll
    for (int r = 0; r < 8; ++r) {
        const int row = rowbase + r + 8 * halfsel;
        if (row < col && tgt[row] == tc) {
            float d2   = fmaxf(sq[row] + sqc - 2.0f * c[r], 0.0f);
            float dist = sqrtf(d2);
            float J    = __logf(row_neg[row] + rnc) + dist;
            J = fmaxf(J, 0.0f);
            s += J * J;
        }
    }
    s += __shfl_xor(s, 1, 32);
    s += __shfl_xor(s, 2, 32);
    s += __shfl_xor(s, 4, 32);
    s += __shfl_xor(s, 8, 32);
    s += __shfl_xor(s, 16, 32);
    if (lane == 0) atomicAdd(out_sum, s);
}

// ===========================================================================
// Generic fallback kernels (runtime D, any multiple of 4; n multiple of 16).
// ===========================================================================
__global__ __launch_bounds__(32) void pass1_rowneg_gen(
        const float* __restrict__ X, const float* __restrict__ sq,
        const int* __restrict__ tgt, float* __restrict__ row_neg,
        int d, int ntiles, int tilesPerChunk) {
    const int rowbase = blockIdx.x * 16;
    const int lane    = threadIdx.x;
    const int l15     = lane & 15;
    const int halfsel = lane >> 4;

    const int ct0 = blockIdx.y * tilesPerChunk;
    int ct1 = ct0 + tilesPerChunk; if (ct1 > ntiles) ct1 = ntiles;

    float sqr[8]; int tr[8];
    #pragma unroll
    for (int r = 0; r < 8; ++r) {
        const int row = rowbase + r + 8 * halfsel;
        sqr[r] = sq[row];
        tr[r]  = tgt[row];
    }

    float acc[8] = {0.f, 0.f, 0.f, 0.f, 0.f, 0.f, 0.f, 0.f};
    for (int ct = ct0; ct < ct1; ++ct) {
        const int colbase = ct * 16;
        v8f c = gram_tile_gen(X, d, rowbase, colbase);
        const float sqc = sq[colbase + l15];
        const int   tc  = tgt[colbase + l15];
        #pragma unroll
        for (int r = 0; r < 8; ++r) {
            float d2   = fmaxf(sqr[r] + sqc - 2.0f * c[r], 0.0f);
            float dist = sqrtf(d2);
            if (tr[r] != tc && dist != 0.0f)
                acc[r] += __expf(MARGIN - dist);
        }
    }

    #pragma unroll
    for (int r = 0; r < 8; ++r) {
        float v = acc[r];
        v += __shfl_xor(v, 1, 32);
        v += __shfl_xor(v, 2, 32);
        v += __shfl_xor(v, 4, 32);
        v += __shfl_xor(v, 8, 32);
        if (l15 == 0) atomicAdd(&row_neg[rowbase + r + 8 * halfsel], v);
    }
}

__global__ __launch_bounds__(32) void pass2_loss_gen(
        const float* __restrict__ X, const float* __restrict__ sq,
        const int* __restrict__ tgt, const float* __restrict__ row_neg,
        float* __restrict__ out_sum, int d) {
    const int rowTile = blockIdx.x;
    const int colTile = blockIdx.y;
    if (colTile < rowTile) return;
    const int rowbase = rowTile * 16;
    const int colbase = colTile * 16;
    const int lane    = threadIdx.x;
    const int l15     = lane & 15;
    const int halfsel = lane >> 4;

    v8f c = gram_tile_gen(X, d, rowbase, colbase);

    const int   col = colbase + l15;
    const float sqc = sq[col];
    const int   tc  = tgt[col];
    const float rnc = row_neg[col];

    float s = 0.0f;
    #pragma unroll
    for (int r = 0; r < 8; ++r) {
        const int row = rowbase + r + 8 * halfsel;
        if (row < col && tgt[row] == tc) {
            float d2   = fmaxf(sq[row] + sqc - 2.0f * c[r], 0.0f);
            float dist = sqrtf(d2);
            float J    = __logf(row_neg[row] + rnc) + dist;
            J = fmaxf(J, 0.0f);
            s += J * J;
        }
    }
    s += __shfl_xor(s, 1, 32);
    s += __shfl_xor(s, 2, 32);
    s += __shfl_xor(s, 4, 32);
    s += __shfl_xor(s, 8, 32);
    s += __shfl_xor(s, 16, 32);
    if (lane == 0) atomicAdd(out_sum, s);
}

// ---------------------------------------------------------------------------
__global__ void finalize_kernel(const float* __restrict__ sum,
                                const unsigned* __restrict__ cnt,
                                float* __restrict__ out) {
    if (threadIdx.x == 0) out[0] = sum[0] / (float)cnt[0];
}

extern "C" void kernel_launch(void* const* d_in, const int* in_sizes, int n_in,
                              void* d_out, int out_size, void* d_ws, size_t ws_size,
                              hipStream_t stream) {
    const float* X   = (const float*)d_in[0];
    const int*   tgt = (const int*)d_in[1];
    const int n = in_sizes[1];
    const int d = in_sizes[0] / n;
    const int ntiles = n / 16;

    float*    sq      = (float*)d_ws;
    float*    row_neg = sq + n;
    float*    sum     = row_neg + n;
    unsigned* cnt     = (unsigned*)(sum + 1);

    hipLaunchKernelGGL(init_ws_kernel, dim3((n + 255) / 256), dim3(256), 0, stream,
                       row_neg, sum, cnt, n);
    hipLaunchKernelGGL(sqnorm_kernel, dim3(n), dim3(32), 0, stream, X, sq, d);
    hipLaunchKernelGGL(count_pos_kernel, dim3((n + 255) / 256), dim3(256), 0, stream,
                       tgt, n, cnt);

    const int CHUNKS = 32;
    const int tilesPerChunk = (ntiles + CHUNKS - 1) / CHUNKS;

    if (d == 128) {
        hipLaunchKernelGGL((pass1_rowneg_fast<128>), dim3(ntiles, CHUNKS), dim3(32),
                           0, stream, X, sq, tgt, row_neg, ntiles, tilesPerChunk);
        hipLaunchKernelGGL((pass2_loss_fast<128>), dim3(ntiles, ntiles), dim3(32),
                           0, stream, X, sq, tgt, row_neg, sum);
    } else {
        hipLaunchKernelGGL(pass1_rowneg_gen, dim3(ntiles, CHUNKS), dim3(32),
                           0, stream, X, sq, tgt, row_neg, d, ntiles, tilesPerChunk);
        hipLaunchKernelGGL(pass2_loss_gen, dim3(ntiles, ntiles), dim3(32),
                           0, stream, X, sq, tgt, row_neg, sum, d);
    }
    hipLaunchKernelGGL(finalize_kernel, dim3(1), dim3(1), 0, stream,
                       sum, cnt, (float*)d_out);
}